// RGAT_11605001634348
// MI455X (gfx1250) — compile-verified
//
#include <hip/hip_runtime.h>

typedef __attribute__((ext_vector_type(2))) float v2f;
typedef __attribute__((ext_vector_type(8))) float v8f;

#define D 128
#define LEAKY_SLOPE 0.01f

// ---- ordered-int encoding for float atomic max (monotonic bijection) ----
__device__ __forceinline__ int f2ord(float f) {
    int i = __float_as_int(f);
    return i >= 0 ? i : (i ^ 0x7fffffff);
}
__device__ __forceinline__ float ord2f(int i) {
    return __int_as_float(i >= 0 ? i : (i ^ 0x7fffffff));
}

// ---------------- init: zero agg, denom; m = INT_MIN -------------------
__global__ void k_init(float* __restrict__ agg, float* __restrict__ denom,
                       int* __restrict__ m_ord, int n_nodes) {
    int total = n_nodes * D;
    for (int i = blockIdx.x * blockDim.x + threadIdx.x; i < total;
         i += gridDim.x * blockDim.x) {
        agg[i] = 0.0f;
        if (i < n_nodes) { denom[i] = 0.0f; m_ord[i] = (int)0x80000000; }
    }
}

// -------------- v[k] = sum_j att_w[2D + j] * w_r[j*D + k] --------------
__global__ void k_attvec(const float* __restrict__ att_w,
                         const float* __restrict__ w_r, float* __restrict__ v) {
    int k = threadIdx.x;              // one block of 128 threads
    float acc = 0.0f;
    for (int j = 0; j < D; ++j) acc += att_w[2 * D + j] * w_r[(size_t)j * D + k];
    v[k] = acc;
}

// -------------- h = x @ w^T : WMMA f32 16x16x4, K=128 -------------------
// block = 256 threads = 8 waves; wave w owns output columns [16w,16w+16)
__global__ void k_proj_h(const float* __restrict__ x, const float* __restrict__ w,
                         float* __restrict__ h, int n_nodes) {
    const int wave = threadIdx.x >> 5;
    const int lane = threadIdx.x & 31;
    const int m16  = lane & 15;
    const int kh   = lane >> 4;                 // 0 or 1 (K half)
    const int row0 = blockIdx.x * 16;
    const int n0   = wave * 16;
    const int arow = row0 + m16 < n_nodes ? row0 + m16 : n_nodes - 1;
    const float* ap = x + (size_t)arow * D + 2 * kh;         // A[m][k..k+1]
    const float* bp = w + (size_t)(n0 + m16) * D + 2 * kh;   // B[k][n]=w[n][k]
    v8f c = {};
#pragma unroll
    for (int k = 0; k < D; k += 4) {
        v2f a = *(const v2f*)(ap + k);
        v2f b = *(const v2f*)(bp + k);
        c = __builtin_amdgcn_wmma_f32_16x16x4_f32(false, a, false, b,
                                                  (short)0, c, false, false);
    }
#pragma unroll
    for (int r = 0; r < 8; ++r) {
        int m = r + 8 * kh;
        if (row0 + m < n_nodes)
            h[(size_t)(row0 + m) * D + n0 + m16] = c[r];
    }
}

// -------------- per-node attention scores s_src, s_dst ------------------
__global__ void k_scores(const float* __restrict__ h, const float* __restrict__ att_w,
                         float* __restrict__ s_src, float* __restrict__ s_dst,
                         int n_nodes) {
    int i = blockIdx.x * blockDim.x + threadIdx.x;
    if (i >= n_nodes) return;
    const float4* hv = (const float4*)(h + (size_t)i * D);
    float a = 0.0f, b = 0.0f;
#pragma unroll 4
    for (int q = 0; q < D / 4; ++q) {
        float4 hq = hv[q];
        a += hq.x * att_w[4 * q]     + hq.y * att_w[4 * q + 1] +
             hq.z * att_w[4 * q + 2] + hq.w * att_w[4 * q + 3];
        b += hq.x * att_w[D + 4 * q]     + hq.y * att_w[D + 4 * q + 1] +
             hq.z * att_w[D + 4 * q + 2] + hq.w * att_w[D + 4 * q + 3];
    }
    s_src[i] = a; s_dst[i] = b;
}

// -------------- edge logits e = leaky(s_src[src]+s_dst[dst]+r_feat.v) ---
// one wave32 per edge; coalesced 32-lane read of the 128-float row
__global__ void k_logits(const float* __restrict__ r_feat, const float* __restrict__ v,
                         const int* __restrict__ src, const int* __restrict__ dst,
                         const float* __restrict__ s_src, const float* __restrict__ s_dst,
                         float* __restrict__ e_buf, int* __restrict__ m_ord, int n_edges) {
    const int wave = threadIdx.x >> 5;
    const int lane = threadIdx.x & 31;
    const int e = blockIdx.x * 8 + wave;
    if (e >= n_edges) return;
    const float* row = r_feat + (size_t)e * D;
    float acc = 0.0f;
#pragma unroll
    for (int q = 0; q < 4; ++q) acc += row[lane + 32 * q] * v[lane + 32 * q];
#pragma unroll
    for (int off = 16; off > 0; off >>= 1) acc += __shfl_down(acc, off, 32);
    if (lane == 0) {
        float val = s_src[src[e]] + s_dst[dst[e]] + acc;
        val = val > 0.0f ? val : LEAKY_SLOPE * val;   // leaky_relu
        e_buf[e] = val;
        atomicMax(&m_ord[dst[e]], f2ord(val));
    }
}

// -------------- ex = exp(e - m[dst]); denom[dst] += ex ------------------
__global__ void k_exp(float* __restrict__ e_buf, const int* __restrict__ m_ord,
                      const int* __restrict__ dst, float* __restrict__ denom,
                      int n_edges) {
    for (int i = blockIdx.x * blockDim.x + threadIdx.x; i < n_edges;
         i += gridDim.x * blockDim.x) {
        int d = dst[i];
        float t = __expf(e_buf[i] - ord2f(m_ord[d]));
        e_buf[i] = t;                          // overwrite with ex
        atomicAdd(&denom[d], t);
    }
}

// -------------- rh (WMMA) fused with alpha*(h[src]+rh) scatter ----------
__global__ void k_msg(const float* __restrict__ r_feat, const float* __restrict__ w_r,
                      const float* __restrict__ h, const int* __restrict__ src,
                      const int* __restrict__ dst, const float* __restrict__ ex,
                      const float* __restrict__ denom, float* __restrict__ agg,
                      int n_edges) {
    const int wave = threadIdx.x >> 5;
    const int lane = threadIdx.x & 31;
    const int m16  = lane & 15;
    const int kh   = lane >> 4;
    const int e0   = blockIdx.x * 16;
    const int n0   = wave * 16;
    const int arow = e0 + m16 < n_edges ? e0 + m16 : n_edges - 1;
    const float* ap = r_feat + (size_t)arow * D + 2 * kh;
    const float* bp = w_r + (size_t)(n0 + m16) * D + 2 * kh;  // B[k][n]=w_r[n][k]
    v8f c = {};
#pragma unroll
    for (int k = 0; k < D; k += 4) {
        v2f a = *(const v2f*)(ap + k);
        v2f b = *(const v2f*)(bp + k);
        c = __builtin_amdgcn_wmma_f32_16x16x4_f32(false, a, false, b,
                                                  (short)0, c, false, false);
    }
#pragma unroll
    for (int r = 0; r < 8; ++r) {
        int m = r + 8 * kh;
        int eidx = e0 + m;
        if (eidx < n_edges) {
            int s = src[eidx], dd = dst[eidx];
            float alpha = ex[eidx] / denom[dd];
            float msg = alpha * (h[(size_t)s * D + n0 + m16] + c[r]);
            atomicAdd(&agg[(size_t)dd * D + n0 + m16], msg);
        }
    }
}

// -------------- out = relu(deg>0 ? agg + h@loop_w : h) ------------------
__global__ void k_out(const float* __restrict__ h, const float* __restrict__ loop_w,
                      const float* __restrict__ agg, const float* __restrict__ denom,
                      float* __restrict__ out, int n_nodes) {
    const int wave = threadIdx.x >> 5;
    const int lane = threadIdx.x & 31;
    const int m16  = lane & 15;
    const int kh   = lane >> 4;
    const int row0 = blockIdx.x * 16;
    const int n0   = wave * 16;
    const int arow = row0 + m16 < n_nodes ? row0 + m16 : n_nodes - 1;
    const float* ap = h + (size_t)arow * D + 2 * kh;
    v8f c = {};
#pragma unroll
    for (int k = 0; k < D; k += 4) {
        v2f a = *(const v2f*)(ap + k);
        v2f b;                                  // B[k][n] = loop_w[k][n0+n]
        b.x = loop_w[(size_t)(k + 2 * kh)     * D + n0 + m16];
        b.y = loop_w[(size_t)(k + 2 * kh + 1) * D + n0 + m16];
        c = __builtin_amdgcn_wmma_f32_16x16x4_f32(false, a, false, b,
                                                  (short)0, c, false, false);
    }
#pragma unroll
    for (int r = 0; r < 8; ++r) {
        int m = r + 8 * kh;
        int node = row0 + m;
        if (node < n_nodes) {
            size_t idx = (size_t)node * D + n0 + m16;
            float base = h[idx];
            float val = denom[node] > 0.0f ? (agg[idx] + c[r]) : base;
            out[idx] = val > 0.0f ? val : 0.0f;
        }
    }
}

extern "C" void kernel_launch(void* const* d_in, const int* in_sizes, int n_in,
                              void* d_out, int out_size, void* d_ws, size_t ws_size,
                              hipStream_t stream) {
    const float* x      = (const float*)d_in[0];
    const float* r_feat = (const float*)d_in[1];
    const int*   src    = (const int*)d_in[2];
    const int*   dst    = (const int*)d_in[3];
    const float* w      = (const float*)d_in[4];
    const float* w_r    = (const float*)d_in[5];
    const float* att_w  = (const float*)d_in[6];
    const float* loop_w = (const float*)d_in[7];
    float* out = (float*)d_out;

    const int n_nodes = in_sizes[0] / D;     // 50000
    const int n_edges = in_sizes[2];         // 640000

    // workspace layout (floats)
    float* ws    = (float*)d_ws;
    float* h     = ws;                                   // n*D
    float* agg   = h + (size_t)n_nodes * D;              // n*D
    float* exbuf = agg + (size_t)n_nodes * D;            // E
    float* s_src = exbuf + n_edges;                      // n
    float* s_dst = s_src + n_nodes;                      // n
    float* denom = s_dst + n_nodes;                      // n
    int*   m_ord = (int*)(denom + n_nodes);              // n
    float* vvec  = (float*)(m_ord + n_nodes);            // D

    const int node_tiles = (n_nodes + 15) / 16;
    const int edge_tiles = (n_edges + 15) / 16;

    k_init<<<512, 256, 0, stream>>>(agg, denom, m_ord, n_nodes);
    k_attvec<<<1, D, 0, stream>>>(att_w, w_r, vvec);
    k_proj_h<<<node_tiles, 256, 0, stream>>>(x, w, h, n_nodes);
    k_scores<<<(n_nodes + 255) / 256, 256, 0, stream>>>(h, att_w, s_src, s_dst, n_nodes);
    k_logits<<<(n_edges + 7) / 8, 256, 0, stream>>>(r_feat, vvec, src, dst,
                                                    s_src, s_dst, exbuf, m_ord, n_edges);
    k_exp<<<1024, 256, 0, stream>>>(exbuf, m_ord, dst, denom, n_edges);
    k_msg<<<edge_tiles, 256, 0, stream>>>(r_feat, w_r, h, src, dst,
                                          exbuf, denom, agg, n_edges);
    k_out<<<node_tiles, 256, 0, stream>>>(h, loop_w, agg, denom, out, n_nodes);
}